// AMCValueNet_45277545234721
// MI455X (gfx1250) — compile-verified
//
#include <hip/hip_runtime.h>
#include <hip/hip_bf16.h>
#include <math.h>

#define NCOAL 320
#define DDIM  512
#define LDST  36   // LDS row stride (floats): 16B-aligned rows, spreads banks

typedef __attribute__((ext_vector_type(2))) float v2f;
typedef __attribute__((ext_vector_type(8))) float v8f;
typedef int v4i __attribute__((vector_size(16)));

#if defined(__has_builtin)
#if __has_builtin(__builtin_amdgcn_global_load_async_to_lds_b128)
#define ASYNC_LDS 1
#endif
#endif

__device__ __forceinline__ v8f wmma_f32_4(v2f a, v2f b, v8f c) {
  // V_WMMA_F32_16X16X4_F32: D = A(16x4 f32) * B(4x16 f32) + C(16x16 f32)
  return __builtin_amdgcn_wmma_f32_16x16x4_f32(
      /*neg_a=*/false, a, /*neg_b=*/false, b,
      /*c_mod=*/(short)0, c, /*reuse_a=*/false, /*reuse_b=*/false);
}

// 16-byte global -> LDS copy: async DMA path (ASYNCcnt) on gfx1250,
// otherwise explicit load + ds_store staging with identical layout.
__device__ __forceinline__ void cp16(float* lds_dst, const float* gsrc) {
#ifdef ASYNC_LDS
  __builtin_amdgcn_global_load_async_to_lds_b128(
      (__attribute__((address_space(1))) v4i*)gsrc,
      (__attribute__((address_space(3))) v4i*)lds_dst, 0, 0);
#else
  *(float4*)lds_dst = *(const float4*)gsrc;
#endif
}

__device__ __forceinline__ void async_drain() {
#ifdef ASYNC_LDS
#if __has_builtin(__builtin_amdgcn_s_wait_asynccnt)
  __builtin_amdgcn_s_wait_asynccnt(0);
#else
  asm volatile("s_wait_asynccnt 0" ::: "memory");
#endif
#endif
}

// ---------------------------------------------------------------------------
// Shared 64x64 NT GEMM tile: C[64,64] += A[64rows,K] * B[64rows,K]^T, K=512.
// Block = 128 threads (4 waves). K staged through LDS in 32-wide chunks with
// coalesced 128-bit copies; WMMA fragments read from LDS (ds_load_b64).
// Wave w owns the 16(M) x 64(N) strip -> 4 v8f accumulators.
// ---------------------------------------------------------------------------
__device__ __forceinline__ void gemm64x64(
    const float* __restrict__ Abase,   // &A[m0][0], row stride DDIM
    const float* __restrict__ Bbase,   // &B[n0][0], row stride DDIM
    float* __restrict__ xs, float* __restrict__ ws,  // LDS, 64*LDST each
    v8f& acc0, v8f& acc1, v8f& acc2, v8f& acc3)
{
  const int t    = threadIdx.x;
  const int lane = t & 31;
  const int wave = t >> 5;
  const int mlo  = lane & 15;            // row within 16-wide tile
  const int koff = (lane >> 4) << 1;     // lanes 16-31 hold K+2,K+3

  for (int kc = 0; kc < DDIM; kc += 32) {
    // Stage A and B 64x32 sub-tiles: 512 float4s each, 4 per thread,
    // 8 consecutive threads cover one contiguous 128B row segment.
    #pragma unroll
    for (int it = 0; it < 4; ++it) {
      const int idx = it * 128 + t;      // 0..511
      const int row = idx >> 3;          // 0..63
      const int c4  = (idx & 7) << 2;    // 0,4,...,28
      cp16(&xs[row * LDST + c4], Abase + (size_t)row * DDIM + kc + c4);
      cp16(&ws[row * LDST + c4], Bbase + (size_t)row * DDIM + kc + c4);
    }
    async_drain();
    __syncthreads();

    const float* ax = &xs[(wave * 16 + mlo) * LDST + koff];
    const float* b0 = &ws[( 0 + mlo) * LDST + koff];
    const float* b1 = &ws[(16 + mlo) * LDST + koff];
    const float* b2 = &ws[(32 + mlo) * LDST + koff];
    const float* b3 = &ws[(48 + mlo) * LDST + koff];
    #pragma unroll
    for (int kk = 0; kk < 32; kk += 4) {
      v2f a   = *(const v2f*)(ax + kk);
      v2f vb0 = *(const v2f*)(b0 + kk);
      v2f vb1 = *(const v2f*)(b1 + kk);
      v2f vb2 = *(const v2f*)(b2 + kk);
      v2f vb3 = *(const v2f*)(b3 + kk);
      acc0 = wmma_f32_4(a, vb0, acc0);
      acc1 = wmma_f32_4(a, vb1, acc1);
      acc2 = wmma_f32_4(a, vb2, acc2);
      acc3 = wmma_f32_4(a, vb3, acc3);
    }
    __syncthreads();
  }
}

// ---------------------------------------------------------------------------
// Kernel 1: fused q/k/v projections.  out = x @ W.T + b  (gridDim.z selects W)
// ---------------------------------------------------------------------------
__global__ __launch_bounds__(128) void qkv_wmma(
    const float* __restrict__ x,
    const float* __restrict__ Wq, const float* __restrict__ bq,
    const float* __restrict__ Wk, const float* __restrict__ bk,
    const float* __restrict__ Wv, const float* __restrict__ bv,
    float* __restrict__ qout, float* __restrict__ kout, float* __restrict__ vout)
{
  __shared__ float xs[64 * LDST];
  __shared__ float ws[64 * LDST];

  const float* W; const float* bias; float* out;
  if (blockIdx.z == 0)      { W = Wq; bias = bq; out = qout; }
  else if (blockIdx.z == 1) { W = Wk; bias = bk; out = kout; }
  else                      { W = Wv; bias = bv; out = vout; }

  const int lane = threadIdx.x & 31;
  const int wave = threadIdx.x >> 5;
  const int mlo  = lane & 15;
  const int mblk = blockIdx.y * 64;
  const int n0   = blockIdx.x * 64;

  v8f acc0 = {}, acc1 = {}, acc2 = {}, acc3 = {};
  gemm64x64(x + (size_t)mblk * DDIM, W + (size_t)n0 * DDIM, xs, ws,
            acc0, acc1, acc2, acc3);

  // C/D layout: VGPR r -> M = r (lanes 0-15) / r+8 (lanes 16-31); N = lane%16
  const int rbase = mblk + wave * 16 + ((lane >> 4) << 3);
  const int c0    = n0 + mlo;
  #pragma unroll
  for (int r = 0; r < 8; ++r) {
    const int row = rbase + r;
    out[(size_t)row * DDIM + c0 +  0] = acc0[r] + bias[c0 +  0];
    out[(size_t)row * DDIM + c0 + 16] = acc1[r] + bias[c0 + 16];
    out[(size_t)row * DDIM + c0 + 32] = acc2[r] + bias[c0 + 32];
    out[(size_t)row * DDIM + c0 + 48] = acc3[r] + bias[c0 + 48];
  }
}

// ---------------------------------------------------------------------------
// Kernel 2: scores = (q @ k.T) / sqrt(D)
// ---------------------------------------------------------------------------
__global__ __launch_bounds__(128) void scores_wmma(
    const float* __restrict__ q, const float* __restrict__ kmat,
    float* __restrict__ sc)
{
  __shared__ float xs[64 * LDST];
  __shared__ float ws[64 * LDST];

  const int lane = threadIdx.x & 31;
  const int wave = threadIdx.x >> 5;
  const int mlo  = lane & 15;
  const int mblk = blockIdx.y * 64;
  const int n0   = blockIdx.x * 64;

  v8f acc0 = {}, acc1 = {}, acc2 = {}, acc3 = {};
  gemm64x64(q + (size_t)mblk * DDIM, kmat + (size_t)n0 * DDIM, xs, ws,
            acc0, acc1, acc2, acc3);

  const float scale = 0.044194173824159216f;  // 1/sqrt(512)
  const int rbase = mblk + wave * 16 + ((lane >> 4) << 3);
  const int c0    = n0 + mlo;
  #pragma unroll
  for (int r = 0; r < 8; ++r) {
    const int row = rbase + r;
    sc[(size_t)row * NCOAL + c0 +  0] = acc0[r] * scale;
    sc[(size_t)row * NCOAL + c0 + 16] = acc1[r] * scale;
    sc[(size_t)row * NCOAL + c0 + 32] = acc2[r] * scale;
    sc[(size_t)row * NCOAL + c0 + 48] = acc3[r] * scale;
  }
}

// ---------------------------------------------------------------------------
// Kernel 3: collapsed prefix-softmax pooling (O(N) per row via scans).
// e_k = exp(s_ik - rowmax); S_j = prefix sum of e; c_j = [j>i]/(j*S_j);
// C = suffix sum of c;  P[k] += e_k * C[max(i,k)+1].
// ---------------------------------------------------------------------------
__global__ __launch_bounds__(512) void prefix_pool(
    const float* __restrict__ sc, float* __restrict__ P)
{
  __shared__ float buf[512];
  __shared__ float ev[512];
  const int t  = threadIdx.x;
  const int i  = blockIdx.x;            // 0..318
  const int NK = NCOAL - 1;             // 319 valid prefix/key positions
  const float* srow = sc + (size_t)i * NCOAL;

  float s = (t < NK) ? srow[t] : -3.4e38f;
  buf[t] = s;
  __syncthreads();
  for (int off = 256; off > 0; off >>= 1) {
    if (t < off) buf[t] = fmaxf(buf[t], buf[t + off]);
    __syncthreads();
  }
  const float rowmax = buf[0];
  __syncthreads();

  const float e = (t < NK) ? expf(s - rowmax) : 0.f;
  ev[t]  = e;
  buf[t] = e;
  __syncthreads();

  // inclusive prefix scan: buf[t] = sum_{t'<=t} e_{t'}  (== S_{j=t+1})
  for (int off = 1; off < 512; off <<= 1) {
    const float val = buf[t];
    const float add = (t >= off) ? buf[t - off] : 0.f;
    __syncthreads();
    buf[t] = val + add;
    __syncthreads();
  }
  const float Sincl = buf[t];
  __syncthreads();

  // c_j stored at index t=j-1; valid prefixes are j in (i, 319]
  const float dj = (t >= i && t < NK) ? 1.f / ((float)(t + 1) * Sincl) : 0.f;
  buf[t] = dj;
  __syncthreads();

  // inclusive suffix scan: buf[t] = sum_{t'>=t} c  (== C_{j=t+1})
  for (int off = 1; off < 512; off <<= 1) {
    const float val = buf[t];
    const float add = (t + off < 512) ? buf[t + off] : 0.f;
    __syncthreads();
    buf[t] = val + add;
    __syncthreads();
  }

  if (t < NK) {
    const int jm = (i > t) ? i : t;     // index of C_{max(i,t)+1}
    atomicAdd(&P[t], ev[t] * buf[jm]);
  }
}

// ---------------------------------------------------------------------------
// Kernel 4: out = Wc[:D] . (P @ v)  +  Wc[D:] . (sum_i x_i)  +  N * bc
// ---------------------------------------------------------------------------
__global__ __launch_bounds__(512) void finalize_k(
    const float* __restrict__ x, const float* __restrict__ vmat,
    const float* __restrict__ P,
    const float* __restrict__ Wc, const float* __restrict__ bc,
    float* __restrict__ out)
{
  __shared__ float red[512];
  const int d = threadIdx.x;
  float feat = 0.f, xsum = 0.f;
  for (int i = 0; i < NCOAL; ++i) {
    feat  = fmaf(P[i], vmat[(size_t)i * DDIM + d], feat);
    xsum += x[(size_t)i * DDIM + d];
  }
  red[d] = feat * Wc[d] + xsum * Wc[DDIM + d];
  __syncthreads();
  for (int s = 256; s > 0; s >>= 1) {
    if (d < s) red[d] += red[d + s];
    __syncthreads();
  }
  if (d == 0) out[0] = red[0] + (float)NCOAL * bc[0];
}

__global__ void zero_vec(float* __restrict__ p, int n) {
  const int t = blockIdx.x * blockDim.x + threadIdx.x;
  if (t < n) p[t] = 0.f;
}

// ---------------------------------------------------------------------------
extern "C" void kernel_launch(void* const* d_in, const int* in_sizes, int n_in,
                              void* d_out, int out_size, void* d_ws, size_t ws_size,
                              hipStream_t stream) {
  (void)in_sizes; (void)n_in; (void)out_size; (void)ws_size;
  const float* x  = (const float*)d_in[0];
  /* d_in[1] = goal (unused by the reference forward) */
  const float* Wq = (const float*)d_in[2];
  const float* bq = (const float*)d_in[3];
  const float* Wk = (const float*)d_in[4];
  const float* bk = (const float*)d_in[5];
  const float* Wv = (const float*)d_in[6];
  const float* bv = (const float*)d_in[7];
  const float* Wc = (const float*)d_in[8];
  const float* bc = (const float*)d_in[9];
  float* out = (float*)d_out;

  float* q    = (float*)d_ws;                 // [320*512]
  float* kmat = q    + NCOAL * DDIM;          // [320*512]
  float* vmat = kmat + NCOAL * DDIM;          // [320*512]
  float* sc   = vmat + NCOAL * DDIM;          // [320*320]
  float* P    = sc   + NCOAL * NCOAL;         // [320]

  zero_vec<<<1, NCOAL, 0, stream>>>(P, NCOAL);
  qkv_wmma<<<dim3(DDIM / 64, NCOAL / 64, 3), 128, 0, stream>>>(
      x, Wq, bq, Wk, bk, Wv, bv, q, kmat, vmat);
  scores_wmma<<<dim3(NCOAL / 64, NCOAL / 64, 1), 128, 0, stream>>>(q, kmat, sc);
  prefix_pool<<<NCOAL - 1, 512, 0, stream>>>(sc, P);
  finalize_k<<<1, 512, 0, stream>>>(x, vmat, P, Wc, bc, out);
}